// MessagePassing_42769284334132
// MI455X (gfx1250) — compile-verified
//
#include <hip/hip_runtime.h>

// MessagePassing (4-direction SCNN scan) for MI455X / gfx1250.
// fp32 math on the matrix units via V_WMMA_F32_16X16X4_F32.

#define CC   128
#define HH   72
#define WW   200
#define KK   9
#define CPAD 132   // LDS row pitch in floats (conflict-free b64 reads)

typedef __attribute__((ext_vector_type(2))) float v2f;
typedef __attribute__((ext_vector_type(8))) float v8f;

// ---------------------------------------------------------------------------
// Weight packing: Apack[dir][k][i2][o] (float2), i2 = i/2.
// A-fragment (16x4 f32): lane l<16 -> (o=m0+l, i0..i0+1), lane l+16 -> (i0+2..3)
// so a wave's A fragment is one b64 load of Apack[(k*64 + i0/2 + (l>>4))*128 + m0+(l&15)].
// ---------------------------------------------------------------------------
__global__ void pack_weights_k(const float* __restrict__ w0, const float* __restrict__ w1,
                               const float* __restrict__ w2, const float* __restrict__ w3,
                               float2* __restrict__ ap) {
  int idx = blockIdx.x * blockDim.x + threadIdx.x;
  const int total = 4 * KK * (CC / 2) * CC;
  if (idx >= total) return;
  int o  = idx & (CC - 1);
  int r  = idx >> 7;
  int i2 = r & 63;  r >>= 6;
  int k  = r % KK;
  int d  = r / KK;
  const float* w = (d == 0) ? w0 : (d == 1) ? w1 : (d == 2) ? w2 : w3;
  int i = i2 * 2;
  float2 v;
  v.x = w[(o * CC + i    ) * KK + k];
  v.y = w[(o * CC + i + 1) * KK + k];
  ap[idx] = v;
}

__global__ void init_cnt_k(unsigned* __restrict__ c) {
  if (threadIdx.x < 64) c[threadIdx.x] = 0u;
}

// Device-scope split barrier among the S blocks of one batch (monotonic counter).
__device__ __forceinline__ void batch_barrier(unsigned* c, unsigned target) {
  __threadfence();          // each wave: own stores visible at device scope
  __syncthreads();          // all waves of this block released their stores
  if (threadIdx.x == 0)
    __hip_atomic_fetch_add(c, 1u, __ATOMIC_RELEASE, __HIP_MEMORY_SCOPE_AGENT);
  if ((threadIdx.x & 31) == 0) {   // one acquire spin per wave (per-CU cache inv)
    while (__hip_atomic_load(c, __ATOMIC_ACQUIRE, __HIP_MEMORY_SCOPE_AGENT) < target)
      __builtin_amdgcn_s_sleep(1);
  }
  __syncthreads();
}

// ---------------------------------------------------------------------------
// One propagation pass. grid = (16 batches, S blocks/batch), block = 512.
// Scan over nPos lines; per step: conv1d(C->C,K=9) over length L on the
// previous (already-updated) line, + relu, added to current line.
//   element addr = batchBase + c*sC + pos*sP + n*sN
// ---------------------------------------------------------------------------
__global__ void prop_pass_k(const float* __restrict__ cur,   // source of "current line" term
                            float* __restrict__ out,
                            const float2* __restrict__ ap,   // packed weights for this dir
                            unsigned* __restrict__ cnt,      // 16 counters (one per batch)
                            int S,
                            long sP, long sN,
                            int L, int NP, int NROW,
                            int nPos, int rev, int doCopy) {
  extern __shared__ float lds[];                  // lds[NROW][CPAD]
  const long sC   = (long)HH * WW;
  const int  b    = blockIdx.x;
  const long base = (long)b * CC * HH * WW;
  unsigned*  c    = cnt + b;
  const int  tid  = threadIdx.x;
  const int  nthr = blockDim.x;
  const int  lane = tid & 31, l15 = lane & 15, lhi = lane >> 4;
  const int  wv   = blockIdx.y * (nthr >> 5) + (tid >> 5);
  const int  totW = S * (nthr >> 5);
  const int  ntile = NP * 8;                      // (Mtile, N-pair) units

  if (doCopy) {                                   // first scan line = input copy
    int pos0 = rev ? (nPos - 1) : 0;
    long p0 = base + (long)pos0 * sP;
    for (int idx = blockIdx.y * nthr + tid; idx < CC * L; idx += S * nthr) {
      int cch = idx / L;
      int n   = idx - cch * L;
      long a  = p0 + (long)cch * sC + (long)n * sN;
      out[a] = cur[a];
    }
  }

  const int fillTot = CC * NROW;
  for (int s = 1; s < nPos; ++s) {
    batch_barrier(c, (unsigned)s * (unsigned)S);  // prev line fully written by all blocks
    int pos  = rev ? (nPos - 1 - s) : s;
    int ppos = rev ? (pos + 1) : (pos - 1);

    // Stage previous line into LDS, transposed with zero halo: lds[n+4][c]
    long pbase = base + (long)ppos * sP;
    for (int idx = tid; idx < fillTot; idx += nthr) {
      int cch = idx / NROW;
      int j   = idx - cch * NROW;
      int n   = j - 4;
      float v = 0.f;
      if (n >= 0 && n < L) v = out[pbase + (long)cch * sC + (long)n * sN];
      lds[j * CPAD + cch] = v;
    }
    __syncthreads();

    long obase = base + (long)pos * sP;
    for (int t = wv; t < ntile; t += totW) {
      int m0 = (t & 7) << 4;          // output-channel tile
      int n0 = (t >> 3) << 5;         // 32-wide position tile pair
      v8f a0 = {};
      v8f a1 = {};
      for (int k = 0; k < KK; ++k) {
        const float*  B0p = &lds[(n0 + l15 + k) * CPAD];
        const float*  B1p = B0p + 16 * CPAD;
        const float2* Ap  = ap + (long)k * (64 * CC) + (m0 + l15);
        #pragma unroll 4
        for (int q = 0; q < 32; ++q) {          // uniform trip count: EXEC all-1s
          int i2 = 2 * q + lhi;                 // lane-dependent addressing only
          v2f A  = *(const v2f*)&Ap[(long)i2 * CC];
          v2f B0 = *(const v2f*)&B0p[2 * i2];
          v2f B1 = *(const v2f*)&B1p[2 * i2];
          a0 = __builtin_amdgcn_wmma_f32_16x16x4_f32(false, A, false, B0,
                                                     (short)0, a0, false, false);
          a1 = __builtin_amdgcn_wmma_f32_16x16x4_f32(false, A, false, B1,
                                                     (short)0, a1, false, false);
        }
      }
      // D layout: VGPR r -> M = m0 + r + 8*(lane>>4), N = n0 + (lane&15) [+16]
      int nA = n0 + l15, nB = nA + 16;
      #pragma unroll
      for (int r = 0; r < 8; ++r) {
        int  m      = m0 + r + (lhi << 3);
        long rowoff = obase + (long)m * sC;
        if (nA < L) {
          long a = rowoff + (long)nA * sN;
          out[a] = cur[a] + fmaxf(a0[r], 0.f);
        }
        if (nB < L) {
          long a = rowoff + (long)nB * sN;
          out[a] = cur[a] + fmaxf(a1[r], 0.f);
        }
      }
    }
    // next-iteration barrier provides the intra-block sync before LDS refill
  }
}

// ---------------------------------------------------------------------------
extern "C" void kernel_launch(void* const* d_in, const int* in_sizes, int n_in,
                              void* d_out, int out_size, void* d_ws, size_t ws_size,
                              hipStream_t stream) {
  const float* x  = (const float*)d_in[0];
  const float* wd = (const float*)d_in[1];
  const float* wu = (const float*)d_in[2];
  const float* wr = (const float*)d_in[3];
  const float* wl = (const float*)d_in[4];
  float* out = (float*)d_out;

  const size_t PKD = (size_t)KK * (CC / 2) * CC;          // float2 per direction
  float2*   ap  = (float2*)d_ws;                          // 4*PKD*8B = 2.36 MB
  unsigned* cnt = (unsigned*)((char*)d_ws + 4 * PKD * sizeof(float2)); // 64 uints

  {
    int total = 4 * KK * (CC / 2) * CC;
    pack_weights_k<<<(total + 255) / 256, 256, 0, stream>>>(wd, wu, wr, wl, ap);
  }
  init_cnt_k<<<1, 64, 0, stream>>>(cnt);

  dim3 blk(512);
  // H-scans (conv over W): L=200 -> NP=7 pairs, NROW=232; nPos=H=72
  {
    dim3 g(16, 4);
    size_t shm = (size_t)232 * CPAD * sizeof(float);
    prop_pass_k<<<g, blk, shm, stream>>>(x,   out, ap + 0 * PKD, cnt +  0, 4,
                                         (long)WW, 1L, WW, 7, 232, HH, 0, 1); // down
    prop_pass_k<<<g, blk, shm, stream>>>(out, out, ap + 1 * PKD, cnt + 16, 4,
                                         (long)WW, 1L, WW, 7, 232, HH, 1, 0); // up
  }
  // W-scans (conv over H): L=72 -> NP=3 pairs, NROW=104; nPos=W=200
  {
    dim3 g(16, 2);
    size_t shm = (size_t)104 * CPAD * sizeof(float);
    prop_pass_k<<<g, blk, shm, stream>>>(out, out, ap + 2 * PKD, cnt + 32, 2,
                                         1L, (long)WW, HH, 3, 104, WW, 0, 0); // right
    prop_pass_k<<<g, blk, shm, stream>>>(out, out, ap + 3 * PKD, cnt + 48, 2,
                                         1L, (long)WW, HH, 3, 104, WW, 1, 0); // left
  }
}